// MultiChannelHiPPO_18202071400697
// MI455X (gfx1250) — compile-verified
//
#include <hip/hip_runtime.h>

typedef __attribute__((ext_vector_type(2))) float v2f;
typedef __attribute__((ext_vector_type(4))) float v4f;
typedef __attribute__((ext_vector_type(8))) float v8f;

#define K_CH   4
#define T_LEN  1024
#define D_DIM  256
#define CHUNK  128    // s-rows staged in LDS per iteration

// c[b,t,k,d] = sum_{s<=t} M[t-s,k] * h[b,s,d]
// Per-batch GEMM: rows r = t*4+k (== output memory order),
// W'[r,s] = M[t-s,k]*(s<=t) generated on the fly from M in LDS.
__global__ __launch_bounds__(256)
void hippo_legt_wmma(const float* __restrict__ h,
                     const float* __restrict__ M,
                     float* __restrict__ out)
{
    __shared__ float Mlds[T_LEN * K_CH];   // 16 KB, [t*4 + k]
    __shared__ float Hs[CHUNK * 32];       // 16 KB, [srel*32 + dc]

    const int tid  = threadIdx.x;
    const int wid  = tid >> 5;       // wave id 0..7
    const int lane = tid & 31;
    const int half = lane >> 4;      // 0: lanes 0-15, 1: lanes 16-31
    const int n    = lane & 15;

    const int d0      = blockIdx.y * 32;
    const int b       = blockIdx.z;
    const int wg_r0   = blockIdx.x * 128;        // 8 waves * 16 rows
    const int wave_r0 = wg_r0 + wid * 16;
    const int wave_t0 = wave_r0 >> 2;            // min t in this wave's tile (mult of 4)

    // Per-lane A-row identity: row m = n, t = r>>2, k = r&3
    const int t_lane  = wave_t0 + (n >> 2);
    const int k_lane  = n & 3;
    const int s_total = (wg_r0 >> 2) + 32;       // WG-wide causal s bound (excl)

    // ---- stage M (T x K = 4096 f32) into LDS, float4-vectorized ----
    {
        const v4f* Msrc = (const v4f*)M;
        v4f*       Mdst = (v4f*)Mlds;
        #pragma unroll
        for (int i = tid; i < (T_LEN * K_CH) / 4; i += 256)
            Mdst[i] = Msrc[i];
    }

    v8f acc0 = {};
    v8f acc1 = {};

    const float* hb  = h + (size_t)b * T_LEN * D_DIM + d0;
    v4f*         Hs4 = (v4f*)Hs;

    for (int cb = 0; cb < s_total; cb += CHUNK) {
        const int cnt = min(CHUNK, s_total - cb);

        __syncthreads();   // prior chunk consumed (also fences Mlds on iter 0)
        // stage h[b, cb..cb+cnt, d0..d0+32) -> Hs, b128 loads/stores
        for (int i = tid; i < cnt * 8; i += 256) {
            const int srel = i >> 3;
            const int c4   = i & 7;
            Hs4[i] = *(const v4f*)(hb + (size_t)(cb + srel) * D_DIM + c4 * 4);
        }
        __syncthreads();

        const int cend = cb + cnt;
        // Wave-uniform bounds -> scalar loop control
        const int send = __builtin_amdgcn_readfirstlane(min(cend, wave_t0 + 4));
        const int bend = __builtin_amdgcn_readfirstlane(min(cend, wave_t0));

        // ---- bulk region: strictly below the diagonal, no causal mask ----
        int aidx = (t_lane - (cb + 2 * half)) * K_CH + k_lane;  // Mlds idx of a0
        int bidx = 2 * half * 32 + n;                           // Hs idx of B0.x
        #pragma unroll 2
        for (int s0 = cb; s0 < bend; s0 += 4) {
            v2f A;
            A.x = Mlds[aidx];
            A.y = Mlds[aidx - K_CH];
            v2f B0, B1;
            B0.x = Hs[bidx];      B0.y = Hs[bidx + 32];
            B1.x = Hs[bidx + 16]; B1.y = Hs[bidx + 48];

            acc0 = __builtin_amdgcn_wmma_f32_16x16x4_f32(
                       false, A, false, B0, (short)0, acc0, false, false);
            acc1 = __builtin_amdgcn_wmma_f32_16x16x4_f32(
                       false, A, false, B1, (short)0, acc1, false, false);

            aidx -= 4 * K_CH;
            bidx += 128;
        }

        // ---- diagonal tail: at most one masked step per wave total ----
        const int ts = max(cb, bend);
        for (int s0 = ts; s0 < send; s0 += 4) {
            const int sa = s0 + 2 * half;
            const int i0 = t_lane - sa;
            const int i1 = i0 - 1;
            const float m0 = Mlds[max(i0, 0) * K_CH + k_lane];
            const float m1 = Mlds[max(i1, 0) * K_CH + k_lane];
            v2f A;
            A.x = (i0 >= 0) ? m0 : 0.0f;
            A.y = (i1 >= 0) ? m1 : 0.0f;

            const int srel = sa - cb;
            v2f B0, B1;
            B0.x = Hs[srel * 32 + n];
            B0.y = Hs[(srel + 1) * 32 + n];
            B1.x = Hs[srel * 32 + 16 + n];
            B1.y = Hs[(srel + 1) * 32 + 16 + n];

            acc0 = __builtin_amdgcn_wmma_f32_16x16x4_f32(
                       false, A, false, B0, (short)0, acc0, false, false);
            acc1 = __builtin_amdgcn_wmma_f32_16x16x4_f32(
                       false, A, false, B1, (short)0, acc1, false, false);
        }
    }

    // ---- store: C/D layout VGPR v -> row (v + 8*half), col n ----
    float* ob = out + ((size_t)b * (T_LEN * K_CH) + wave_r0) * D_DIM + d0;
    #pragma unroll
    for (int v = 0; v < 8; ++v) {
        const int m = v + half * 8;
        ob[(size_t)m * D_DIM + n]      = acc0[v];
        ob[(size_t)m * D_DIM + 16 + n] = acc1[v];
    }
}

extern "C" void kernel_launch(void* const* d_in, const int* in_sizes, int n_in,
                              void* d_out, int out_size, void* d_ws, size_t ws_size,
                              hipStream_t stream) {
    const float* h = (const float*)d_in[0];   // (B, T, D) f32
    const float* M = (const float*)d_in[1];   // (T, K) f32
    float* out = (float*)d_out;               // (B, T, K, D) f32

    dim3 grid(32, 8, 2);   // r-chunks x d-tiles x batch
    dim3 block(256);       // 8 waves (wave32)
    hipLaunchKernelGGL(hippo_legt_wmma, grid, block, 0, stream, h, M, out);
}